// Benes_75067438399652
// MI455X (gfx1250) — compile-verified
//
#include <hip/hip_runtime.h>

// ---------------------------------------------------------------------------
// Benes network (N=4096, 12 butterfly stages) on MI455X / gfx1250.
//
// 12 sparse butterfly stages folded into 3 passes of dense 16x16 GEMMs
// (radix-16 decomposition) executed with V_WMMA_F32_16X16X4_F32.
// All 12 stages fused in one trip through HBM (128 MB -> ~5.5us at 23.3TB/s).
// Slab lives in LDS, batch-major [16 rows][4097 words] = 256 KB; the
// global<->LDS tile movement is done by the Tensor Data Mover (TDM) in
// iterate mode (16 x contiguous 4096-f32 rows, LDS pitch 4097), tracked by
// TENSORcnt, so the 32 compute waves never touch the bulk copy.
// ---------------------------------------------------------------------------

typedef float    v2f  __attribute__((ext_vector_type(2)));
typedef float    v8f  __attribute__((ext_vector_type(8)));
typedef uint32_t u32x4 __attribute__((ext_vector_type(4)));
typedef uint32_t u32x8 __attribute__((ext_vector_type(8)));

#define BN       4096            // element dimension
#define BNNZ     (2*BN)          // nonzeros per stage layer
#define ROWPITCH 4097            // LDS words per batch row (== 1 mod 64)

__device__ __forceinline__ int elem_index(int g, int id, int r) {
    if (g == 0) return (id << 4) | r;
    if (g == 1) return ((id >> 4) << 8) | (r << 4) | (id & 15);
    return (r << 8) | id;
}

// ---------------------------------------------------------------------------
// Pass A: build 3*256 dense 16x16 matrices, each the product of 4 sparse
// butterfly stages (closed form of the Python index constructor):
//   stage j, stride c=2^j, element i: beta=(i>>j)&1, k=i&(c-1),
//   vb=(i>>(j+1))<<(j+2):
//     coeff_on_low_partner  = V[vb + beta*c + k]
//     coeff_on_high_partner = V[vb + 2c + beta*c + k]
// ---------------------------------------------------------------------------
__global__ void build_mats_kernel(const float* __restrict__ values,
                                  float* __restrict__ mats) {
    const int bid = blockIdx.x;       // 0..767  = g*256 + id
    const int g   = bid >> 8;
    const int id  = bid & 255;
    const int col = threadIdx.x & 15; // input-row index of the matrix
    const int r   = threadIdx.x >> 4; // output-row index

    __shared__ float T[16][17];
    T[r][col] = (r == col) ? 1.0f : 0.0f;
    __syncthreads();

    #pragma unroll
    for (int s = 0; s < 4; ++s) {
        const int j  = g * 4 + s;
        const int c  = 1 << j;
        const int cl = 1 << s;
        const int i  = elem_index(g, id, r);
        const int beta = (r >> s) & 1;
        const int k  = i & (c - 1);
        const int vb = (i >> (j + 1)) << (j + 2);
        const float* V = values + j * BNNZ;
        const float clow  = V[vb + beta * c + k];
        const float chigh = V[vb + 2 * c + beta * c + k];
        const float a = T[r & ~cl][col];
        const float b = T[r |  cl][col];
        __syncthreads();
        T[r][col] = clow * a + chigh * b;
        __syncthreads();
    }
    mats[bid * 256 + r * 16 + col] = T[r][col];
}

// ---------------------------------------------------------------------------
// TDM descriptor (ISA cdna5 ch.8): 16 iterations of a 1-D tile of 4096 f32.
//   group0: count=1 | lds_addr | global_addr(57b) | type=2
//   group1: data_size=4B, iterate_enable, tensor_dim0=4096, tile_dim0=4096,
//           tensor_dim0_stride=4096
//   group2 (iterate mode): lds_addr_increment=ROWPITCH,
//           global_addr_increment=4096, iterate_count=15 (=> 16 rows)
// Issued wave-uniform from SGPR tuples; tracked with TENSORcnt.
// ---------------------------------------------------------------------------
__device__ __forceinline__ void tdm_tile(uint64_t gaddr, uint32_t lds_byte,
                                         bool store) {
    u32x4 g0;
    g0[0] = 1u;                                             // count=1 (valid)
    g0[1] = lds_byte;                                       // lds_addr
    g0[2] = (uint32_t)(gaddr & 0xFFFFFFFFull);              // global_addr lo
    g0[3] = (uint32_t)((gaddr >> 32) & 0x01FFFFFFull)       // global_addr hi
          | (2u << 30);                                     // type = image
    u32x8 g1;
    g1[0] = (2u << 16) | (1u << 19);       // data_size=4B, iterate_enable=1
    g1[1] = (uint32_t)(BN & 0xFFFF) << 16; // tensor_dim0[15:0]
    g1[2] = (uint32_t)(BN >> 16) | (16u << 16); // dim0 hi | tensor_dim1 lo
    g1[3] = (uint32_t)BN << 16;            // tile_dim0 = 4096
    g1[4] = 0u;                            // tile_dim1=0 (1-D tile)
    g1[5] = (uint32_t)BN;                  // tensor_dim0_stride lo32
    g1[6] = 0u;
    g1[7] = 0u;
    u32x4 g2;
    g2[0] = 0u;
    g2[1] = (uint32_t)ROWPITCH;            // lds_addr_increment (elem units)
    g2[2] = (uint32_t)BN;                  // global_addr_increment lo32
    g2[3] = 15u << 16;                     // iterate_count = 15 -> 16 iters
    u32x4 g3 = (u32x4)(0u);

    if (store) {
        asm volatile("tensor_store_from_lds %0, %1, %2, %3"
                     :: "s"(g0), "s"(g1), "s"(g2), "s"(g3) : "memory");
    } else {
        asm volatile("tensor_load_to_lds %0, %1, %2, %3"
                     :: "s"(g0), "s"(g1), "s"(g2), "s"(g3) : "memory");
    }
}

// ---------------------------------------------------------------------------
// One WMMA pass, specialized on group G so every LDS address is base+const.
//   slab word addr for (batch L, element i) = L*ROWPITCH + i
//   G=0: i = id*16 + r           (row stride 1)
//   G=1: i = hi*256 + r*16 + lo  (row stride 16)
//   G=2: i = r*256 + id          (row stride 256)
// Fragment layouts (ISA 7.12.2):
//   A 16x4 : lanes 0-15 hold A[L][0..1], lanes 16-31 hold A[L][2..3]
//   B 4x16 : VGPR q <-> K rows {q, q+2};  C/D: VGPR m <-> M rows {m, m+8}
// ---------------------------------------------------------------------------
template <int G>
__device__ __forceinline__ void wmma_pass(float* __restrict__ slab,
                                          const float* __restrict__ mats,
                                          int wave, int L, int half) {
    constexpr int ESTRIDE = (G == 0) ? 1 : (G == 1) ? 16 : 256;

    #pragma unroll 2
    for (int kblk = 0; kblk < 8; ++kblk) {
        const int id = wave + (kblk << 5);                  // 0..255
        int e0;                                             // element at r=0
        if (G == 0)      e0 = id << 4;
        else if (G == 1) e0 = ((id >> 4) << 8) | (id & 15);
        else             e0 = id;
        const int abase = L * ROWPITCH + e0;                // LDS word base
        const float* __restrict__ M = mats + ((G * 256 + id) << 8);
        __builtin_prefetch(M, 0, 0);                        // global_prefetch_b8

        v8f acc = {};
        #pragma unroll
        for (int q = 0; q < 4; ++q) {
            const v2f a = *(const v2f*)(M + (L << 4) + (q << 2) + (half << 1));
            const int r0 = (q << 2) + (half << 1);
            v2f b;
            b.x = slab[abase + (r0    ) * ESTRIDE];
            b.y = slab[abase + (r0 + 1) * ESTRIDE];
            acc = __builtin_amdgcn_wmma_f32_16x16x4_f32(
                    false, a, false, b, (short)0, acc, false, false);
        }
        #pragma unroll
        for (int mrow = 0; mrow < 8; ++mrow) {
            slab[abase + (mrow + (half << 3)) * ESTRIDE] = acc[mrow];
        }
    }
}

// ---------------------------------------------------------------------------
// Main fused kernel: one workgroup (1024 thr = 32 wave32) per 16-row batch
// tile. TDM loads the tile, 3 WMMA passes transform it in LDS, TDM stores it.
// ---------------------------------------------------------------------------
__global__ void benes_wmma_kernel(const float* __restrict__ x,
                                  const float* __restrict__ mats,
                                  float* __restrict__ out) {
    extern __shared__ float slab[];   // 16 * ROWPITCH floats (256 KB)

    const int t  = threadIdx.x;       // 0..1023
    const int bt = blockIdx.x;        // batch tile (16 rows)
    const uint32_t lds_byte = (uint32_t)(uintptr_t)slab; // low 32b == LDS offset

    // TDM DMA: x[bt*16 .. bt*16+15][0..4095] -> slab rows (pitch ROWPITCH)
    if (t < 32) {
        tdm_tile((uint64_t)(uintptr_t)(x + (size_t)bt * 16 * BN), lds_byte,
                 /*store=*/false);
        __builtin_amdgcn_s_wait_tensorcnt(0);
    }
    __syncthreads();

    const int wave = t >> 5;          // 0..31 (wave32)
    const int lane = t & 31;
    const int L    = lane & 15;
    const int half = lane >> 4;

    wmma_pass<0>(slab, mats, wave, L, half);
    __syncthreads();
    wmma_pass<1>(slab, mats, wave, L, half);
    __syncthreads();
    wmma_pass<2>(slab, mats, wave, L, half);
    __syncthreads();

    // TDM DMA: slab rows -> out[bt*16 .. bt*16+15][0..4095]
    if (t < 32) {
        tdm_tile((uint64_t)(uintptr_t)(out + (size_t)bt * 16 * BN), lds_byte,
                 /*store=*/true);
        __builtin_amdgcn_s_wait_tensorcnt(0);
    }
}

extern "C" void kernel_launch(void* const* d_in, const int* in_sizes, int n_in,
                              void* d_out, int out_size, void* d_ws, size_t ws_size,
                              hipStream_t stream) {
    const float* x      = (const float*)d_in[0];   // [4096, 4096] f32
    const float* values = (const float*)d_in[1];   // [12, 8192]   f32
    // d_in[2], d_in[3] (idx_in/idx_out) unused: indices are analytic.
    float* outp = (float*)d_out;
    float* mats = (float*)d_ws;                    // 3*256*256 f32 = 768 KB

    build_mats_kernel<<<768, 256, 0, stream>>>(values, mats);

    const size_t lds_bytes = (size_t)16 * ROWPITCH * sizeof(float); // 256 KB
    (void)hipFuncSetAttribute((const void*)benes_wmma_kernel,
                              hipFuncAttributeMaxDynamicSharedMemorySize,
                              (int)lds_bytes);
    benes_wmma_kernel<<<256, 1024, lds_bytes, stream>>>(x, mats, outp);
}